// VQEmbeddingEMA_58926951301459
// MI455X (gfx1250) — compile-verified
//
#include <hip/hip_runtime.h>
#include <math.h>

// Problem constants (from reference): N=4, M=1024, D=128, B=32, H=W=32
#define NN 4
#define MM 1024
#define DD 128
#define BB 32
#define HWSZ 1024          // H*W
#define CC 512             // N*D
#define LL 32768           // B*H*W
#define TOTAL_ELEMS 16777216  // N*L*D == B*C*H*W
#define TILE_L 64
#define XSTRIDE 132        // padded LDS row stride (floats); 528B = 33*16 -> 16B aligned rows

typedef __attribute__((ext_vector_type(2))) float v2f;
typedef __attribute__((ext_vector_type(8))) float v8f;

// pointer types matching the async-load builtin's prototype
typedef int v4i_vs __attribute__((vector_size(16)));
typedef __attribute__((address_space(1))) v4i_vs* g_v4i_ptr;   // global
typedef __attribute__((address_space(3))) v4i_vs* l_v4i_ptr;   // LDS

// ---------------------------------------------------------------------------
// CDNA5 async global->LDS copy (ASYNCcnt path) with graceful fallbacks
// ---------------------------------------------------------------------------
__device__ __forceinline__ void async_ld_b128(float* lds, const float* gp) {
#if defined(__has_builtin) && __has_builtin(__builtin_amdgcn_global_load_async_to_lds_b128)
    __builtin_amdgcn_global_load_async_to_lds_b128(
        (g_v4i_ptr)(void*)gp,       // global source (AS1), const dropped
        (l_v4i_ptr)(void*)lds,      // LDS destination (AS3)
        0, 0);
#else
    unsigned lo = (unsigned)(size_t)(__attribute__((address_space(3))) char*)lds;
    unsigned long long ga = (unsigned long long)gp;
    asm volatile("global_load_async_to_lds_b128 %0, %1, off"
                 :: "v"(lo), "v"(ga) : "memory");
#endif
}

__device__ __forceinline__ void wait_async0() {
#if defined(__has_builtin) && __has_builtin(__builtin_amdgcn_s_wait_asynccnt)
    __builtin_amdgcn_s_wait_asynccnt(0);
#else
    asm volatile("s_wait_asynccnt 0x0" ::: "memory");
#endif
}

// ---------------------------------------------------------------------------
// Kernel 0: zero histogram counts and loss accumulator in workspace
// ---------------------------------------------------------------------------
__global__ void vq_zero_kernel(int* __restrict__ counts, float* __restrict__ loss_acc) {
    int id = blockIdx.x * blockDim.x + threadIdx.x;
    if (id < NN * MM) counts[id] = 0;
    if (id == 0) *loss_acc = 0.0f;
}

// ---------------------------------------------------------------------------
// Kernel 1: e2[n][m] = sum_d embedding[n][m][d]^2
// ---------------------------------------------------------------------------
__global__ void vq_e2_kernel(const float* __restrict__ emb, float* __restrict__ e2) {
    int id = blockIdx.x * blockDim.x + threadIdx.x;
    if (id < NN * MM) {
        const float* row = emb + (size_t)id * DD;
        float s = 0.0f;
        #pragma unroll 8
        for (int d = 0; d < DD; ++d) s += row[d] * row[d];
        e2[id] = s;
    }
}

// ---------------------------------------------------------------------------
// Kernel 2: WMMA distance + argmin, with async double-buffered codebook chunks.
// One workgroup = 128 threads = 4 wave32, handles 64 rows of x_flat[n].
// Each wave owns a 16-row subtile; loops over all 1024 codes in chunks of 16:
//   - chunk c+1 is copied global->LDS via GLOBAL_LOAD_ASYNC_TO_LDS_B128
//     while the 32x V_WMMA_F32_16X16X4_F32 burst runs on chunk c.
// Score = e2[m] - 2*dot (the ||x||^2 term is row-constant, dropped for argmin).
// ---------------------------------------------------------------------------
__global__ __launch_bounds__(128) void vq_argmin_kernel(
    const float* __restrict__ x,       // (B, C, H, W)
    const float* __restrict__ emb,     // (N, M, D)
    const float* __restrict__ e2,      // (N, M)
    int*  __restrict__ indices,        // (N, L)
    int*  __restrict__ counts)         // (N, M)
{
    __shared__ __align__(16) float xs[TILE_L * XSTRIDE];   // x tile (64 x 128)
    __shared__ __align__(16) float es[2][16 * XSTRIDE];    // ping-pong emb chunks

    const int tid  = threadIdx.x;
    const int wg   = blockIdx.x;
    const int n    = wg / (LL / TILE_L);
    const int tile = wg % (LL / TILE_L);
    const int row0 = tile * TILE_L;
    const float* embn = emb + (size_t)n * MM * DD;

    // ---- Stage x tile into LDS (coalesced along L within fixed d) ----
    // x_flat[n][l][d] = x[b][n*DD + d][h][w], l = b*HW + hw
    {
        const int b   = row0 >> 10;         // TILE_L=64 divides HW -> single b per tile
        const int hw0 = row0 & 1023;
        #pragma unroll
        for (int i = 0; i < (TILE_L * DD) / 128; ++i) {   // 64 iterations
            int idx = i * 128 + tid;
            int r = idx & (TILE_L - 1);
            int d = idx / TILE_L;
            xs[r * XSTRIDE + d] =
                x[((size_t)(b * CC + n * DD + d)) * HWSZ + (hw0 + r)];
        }
    }

    // ---- Async-prefetch chunk 0 into buffer 0 ----
    // 16 codes x 128 floats = 512 x b128; 4 issues per thread.
    {
        const float* src = embn;            // chunk 0
        #pragma unroll
        for (int i = 0; i < 4; ++i) {
            int e  = i * 128 + tid;         // 0..511
            int c  = e >> 5;                // code row 0..15
            int d4 = e & 31;                // float4 index within row
            async_ld_b128(&es[0][c * XSTRIDE + d4 * 4], src + c * DD + d4 * 4);
        }
    }
    wait_async0();
    __syncthreads();

    const int lane = tid & 31;
    const int wave = tid >> 5;          // 0..3
    const int rlo  = lane & 15;         // A row / B col / C col (mod 16)
    const int hi   = lane >> 4;         // half select
    const float* xrow = &xs[(wave * 16 + rlo) * XSTRIDE + 2 * hi];
    const int    eoff = rlo * XSTRIDE + 2 * hi;

    float minval[8];
    int   minidx[8];
    #pragma unroll
    for (int v = 0; v < 8; ++v) { minval[v] = 3.4e38f; minidx[v] = 0; }

    for (int chunk = 0; chunk < MM / 16; ++chunk) {
        const int buf = chunk & 1;

        // ---- Kick off async copy of the NEXT chunk into the other buffer ----
        if (chunk + 1 < MM / 16) {
            const float* src = embn + (size_t)(chunk + 1) * 16 * DD;
            #pragma unroll
            for (int i = 0; i < 4; ++i) {
                int e  = i * 128 + tid;
                int c  = e >> 5;
                int d4 = e & 31;
                async_ld_b128(&es[buf ^ 1][c * XSTRIDE + d4 * 4],
                              src + c * DD + d4 * 4);
            }
        }

        // per-lane e2 for this lane's code column (L2-resident, overlaps WMMAs)
        const float ecol = e2[n * MM + chunk * 16 + rlo];

        // ---- 16x16 dot tile over K=128 via 32 f32 WMMAs ----
        const float* erow = &es[buf][eoff];
        v8f acc = {};
        #pragma unroll
        for (int k = 0; k < 32; ++k) {
            v2f a = *(const v2f*)(xrow + 4 * k);
            v2f b = *(const v2f*)(erow + 4 * k);
            acc = __builtin_amdgcn_wmma_f32_16x16x4_f32(
                false, a, false, b, (short)0, acc, false, false);
        }

        // C/D layout: slot v, this lane -> row (v + 8*hi), col rlo
        const int code = chunk * 16 + rlo;
        #pragma unroll
        for (int v = 0; v < 8; ++v) {
            float score = ecol - 2.0f * acc[v];
            if (score < minval[v]) { minval[v] = score; minidx[v] = code; }
        }

        // next chunk's data must be in LDS (all waves) before it is consumed;
        // barrier also protects buf from being overwritten while still in use.
        wait_async0();
        __syncthreads();
    }

    // ---- Cross-lane argmin reduce within each 16-lane half ----
    #pragma unroll
    for (int off = 8; off >= 1; off >>= 1) {
        #pragma unroll
        for (int v = 0; v < 8; ++v) {
            float oval = __shfl_xor(minval[v], off, 32);
            int   oidx = __shfl_xor(minidx[v], off, 32);
            if (oval < minval[v] || (oval == minval[v] && oidx < minidx[v])) {
                minval[v] = oval; minidx[v] = oidx;
            }
        }
    }

    if (rlo == 0) {
        #pragma unroll
        for (int v = 0; v < 8; ++v) {
            int row  = wave * 16 + hi * 8 + v;
            int l    = row0 + row;
            int code = minidx[v];
            indices[n * LL + l] = code;
            atomicAdd(&counts[n * MM + code], 1);
        }
    }
}

// ---------------------------------------------------------------------------
// Kernel 3: gather quantized rows, write straight-through output, accumulate
// commitment squared error. Coalesced along L (the HW axis of x/out).
// ---------------------------------------------------------------------------
__global__ __launch_bounds__(256) void vq_gather_kernel(
    const float* __restrict__ x,
    const float* __restrict__ emb,
    const int*  __restrict__ indices,
    float* __restrict__ out,
    float* __restrict__ loss_acc)
{
    __shared__ float red[256];
    int id = blockIdx.x * 256 + threadIdx.x;   // 0 .. 2^24-1
    int d  = id >> 17;                         // / (N*L)
    int p  = id & 131071;                      // (n, l)
    int n  = p >> 15;
    int l  = p & 32767;
    int b  = l >> 10;
    int hw = l & 1023;

    int code = indices[p];
    float q  = emb[((size_t)(n * MM + code)) * DD + d];
    size_t xi = ((size_t)(b * CC + n * DD + d)) * HWSZ + hw;
    float xv = x[xi];
    out[xi] = xv + (q - xv);                   // straight-through arithmetic
    float diff = xv - q;

    red[threadIdx.x] = diff * diff;
    __syncthreads();
    #pragma unroll
    for (int s = 128; s > 0; s >>= 1) {
        if (threadIdx.x < s) red[threadIdx.x] += red[threadIdx.x + s];
        __syncthreads();
    }
    if (threadIdx.x == 0) atomicAdd(loss_acc, red[0]);
}

// ---------------------------------------------------------------------------
// Kernel 4: loss + perplexity scalars
// ---------------------------------------------------------------------------
__global__ __launch_bounds__(256) void vq_finalize_kernel(
    const int* __restrict__ counts,
    const float* __restrict__ loss_acc,
    float* __restrict__ out_scalars)
{
    __shared__ float red[256];
    __shared__ float perp_sum_s;
    if (threadIdx.x == 0) perp_sum_s = 0.0f;
    __syncthreads();

    for (int n = 0; n < NN; ++n) {
        float s = 0.0f;
        for (int m = threadIdx.x; m < MM; m += 256) {
            float pm = (float)counts[n * MM + m] / (float)LL;
            s += pm * logf(pm + 1e-10f);
        }
        red[threadIdx.x] = s;
        __syncthreads();
        #pragma unroll
        for (int st = 128; st > 0; st >>= 1) {
            if (threadIdx.x < st) red[threadIdx.x] += red[threadIdx.x + st];
            __syncthreads();
        }
        if (threadIdx.x == 0) perp_sum_s += expf(-red[0]);
        __syncthreads();
    }
    if (threadIdx.x == 0) {
        out_scalars[0] = 0.25f * loss_acc[0] / (float)TOTAL_ELEMS;  // loss
        out_scalars[1] = perp_sum_s;                                 // sum perplexity
    }
}

// ---------------------------------------------------------------------------
extern "C" void kernel_launch(void* const* d_in, const int* in_sizes, int n_in,
                              void* d_out, int out_size, void* d_ws, size_t ws_size,
                              hipStream_t stream) {
    const float* x   = (const float*)d_in[0];   // (B, C, H, W) f32
    const float* emb = (const float*)d_in[1];   // (N, M, D)   f32
    float* out = (float*)d_out;                 // out (16M) + loss + perplexity

    char* ws = (char*)d_ws;
    float* e2       = (float*)ws;                                     // 4096 f32
    int*   indices  = (int*)(ws + 16384);                             // 131072 i32
    int*   counts   = (int*)(ws + 16384 + 524288);                    // 4096 i32
    float* loss_acc = (float*)(ws + 16384 + 524288 + 16384);          // 1 f32

    vq_zero_kernel<<<16, 256, 0, stream>>>(counts, loss_acc);
    vq_e2_kernel<<<16, 256, 0, stream>>>(emb, e2);
    vq_argmin_kernel<<<NN * (LL / TILE_L), 128, 0, stream>>>(x, emb, e2, indices, counts);
    vq_gather_kernel<<<TOTAL_ELEMS / 256, 256, 0, stream>>>(x, emb, indices, out, loss_acc);
    vq_finalize_kernel<<<1, 256, 0, stream>>>(counts, loss_acc, out + TOTAL_ELEMS);
}